// GCNLayer_10969346474530
// MI455X (gfx1250) — compile-verified
//
#include <hip/hip_runtime.h>

typedef __attribute__((ext_vector_type(2))) float v2f;
typedef __attribute__((ext_vector_type(8))) float v8f;

#define K_DIM   256
#define OUT_DIM 32

// ---------------------------------------------------------------------------
// Kernel 1: hp = h @ W   via V_WMMA_F32_16X16X4_F32 (fp32 matrix cores)
// One wave -> one 16x16 output tile. 8 waves/block = 4 row-tiles x 2 col-tiles.
// ---------------------------------------------------------------------------
__global__ __launch_bounds__(256) void gcn_gemm_wmma_f32(
    const float* __restrict__ h,   // [N, 256] row-major
    const float* __restrict__ w,   // [256, 32] row-major
    float* __restrict__ hp,        // [N, 32]
    int nRowTiles) {
  const int lane    = threadIdx.x & 31;
  const int wave    = threadIdx.x >> 5;
  const int rowTile = blockIdx.x * 4 + (wave >> 1);
  if (rowTile >= nRowTiles) return;          // wave-uniform exit, EXEC stays all-1s
  const int colTile = wave & 1;
  const int half    = lane >> 4;             // 0: K pair {k,k+1}; 1: {k+2,k+3}
  const int lm      = lane & 15;

  const float* hrow = h + (size_t)(rowTile * 16 + lm) * K_DIM; // A: row = M = lm
  const float* wcol = w + (size_t)(colTile * 16 + lm);         // B: col = N = lm

  v8f acc = {};
#pragma unroll 8
  for (int k = 0; k < K_DIM; k += 4) {
    // A frag: lane holds h[row][k + half*2], h[row][k + half*2 + 1]
    v2f a = *(const v2f*)(hrow + k + half * 2);
    // B frag: lane holds w[k + half*2][col], w[k + half*2 + 1][col]
    v2f b;
    b.x = wcol[(size_t)(k + half * 2) * OUT_DIM];
    b.y = wcol[(size_t)(k + half * 2 + 1) * OUT_DIM];
    acc = __builtin_amdgcn_wmma_f32_16x16x4_f32(
        /*neg_a=*/false, a, /*neg_b=*/false, b,
        /*c_mod=*/(short)0, acc, /*reuse_a=*/false, /*reuse_b=*/false);
  }

  // D layout: VGPR i -> row (i + half*8), col lm
  float* outBase = hp + (size_t)rowTile * 16 * OUT_DIM + colTile * 16 + lm;
#pragma unroll
  for (int i = 0; i < 8; ++i) {
    outBase[(size_t)(i + half * 8) * OUT_DIM] = acc[i];
  }
}

// ---------------------------------------------------------------------------
// Kernel 2: zero the accumulator (d_out doubles as segment-sum buffer)
// ---------------------------------------------------------------------------
__global__ void gcn_zero_f32(float* __restrict__ p, int n) {
  int i = blockIdx.x * blockDim.x + threadIdx.x;
  if (i < n) p[i] = 0.0f;
}

// ---------------------------------------------------------------------------
// Kernel 3: edge scatter.  8 lanes per edge, float4 slice per lane:
//   accum[dst] += hp[src] * norm[src]
// Gather = one 128B line per edge; scatter = global_atomic_add_f32.
// ---------------------------------------------------------------------------
__global__ __launch_bounds__(256) void gcn_edge_scatter(
    const float* __restrict__ hp,    // [N, 32]
    const float* __restrict__ norm,  // [N]
    const int*   __restrict__ src,
    const int*   __restrict__ dst,
    float* __restrict__ accum,       // [N, 32]
    int nEdges) {
  long long t = (long long)blockIdx.x * blockDim.x + threadIdx.x;
  int e  = (int)(t >> 3);
  int c4 = (int)(t & 7) * 4;
  if (e >= nEdges) return;

  int s = src[e];
  int d = dst[e];
  float ns = norm[s];
  float4 v = *(const float4*)(hp + (size_t)s * OUT_DIM + c4);
  float* ap = accum + (size_t)d * OUT_DIM + c4;
  atomicAdd(ap + 0, v.x * ns);
  atomicAdd(ap + 1, v.y * ns);
  atomicAdd(ap + 2, v.z * ns);
  atomicAdd(ap + 3, v.w * ns);
}

// ---------------------------------------------------------------------------
// Kernel 4: out = relu(accum * norm[row] + bias[col]), in-place on d_out
// ---------------------------------------------------------------------------
__global__ void gcn_finalize(float* __restrict__ out,
                             const float* __restrict__ norm,
                             const float* __restrict__ bias,
                             int nElems) {
  int i = blockIdx.x * blockDim.x + threadIdx.x;
  if (i >= nElems) return;
  int row = i >> 5;       // OUT_DIM == 32
  int col = i & 31;
  float v = out[i] * norm[row] + bias[col];
  out[i] = fmaxf(v, 0.0f);
}

// ---------------------------------------------------------------------------
// Launcher. Inputs (setup_inputs order): h, weight, bias, norm, src, dst
// ---------------------------------------------------------------------------
extern "C" void kernel_launch(void* const* d_in, const int* in_sizes, int n_in,
                              void* d_out, int out_size, void* d_ws, size_t ws_size,
                              hipStream_t stream) {
  const float* h    = (const float*)d_in[0];
  const float* w    = (const float*)d_in[1];
  const float* bias = (const float*)d_in[2];
  const float* norm = (const float*)d_in[3];
  const int*   src  = (const int*)d_in[4];
  const int*   dst  = (const int*)d_in[5];

  const int N = in_sizes[3];        // 50000 (norm has one entry per node)
  const int E = in_sizes[4];        // 1600000 edges

  float* hp    = (float*)d_ws;      // [N,32] projected features (6.4 MB)
  float* accum = (float*)d_out;     // d_out doubles as the segment-sum buffer

  const int nOut      = N * OUT_DIM;
  const int nRowTiles = (N + 15) / 16;   // 3125, exact

  // 1) zero accumulator (d_out) for deterministic segment-sum
  gcn_zero_f32<<<(nOut + 255) / 256, 256, 0, stream>>>(accum, nOut);

  // 2) dense projection via fp32 WMMA
  gcn_gemm_wmma_f32<<<(nRowTiles + 3) / 4, 256, 0, stream>>>(h, w, hp, nRowTiles);

  // 3) edge gather * norm[src], atomic scatter-add into accum
  long long edgeThreads = (long long)E * 8;
  int edgeBlocks = (int)((edgeThreads + 255) / 256);
  gcn_edge_scatter<<<edgeBlocks, 256, 0, stream>>>(hp, norm, src, dst, accum, E);

  // 4) accum * norm[dstnode] + bias, relu — in place on d_out
  gcn_finalize<<<(nOut + 255) / 256, 256, 0, stream>>>(accum, norm, bias, nOut);
}